// RandomProjectionQuantizer_91182155694321
// MI455X (gfx1250) — compile-verified
//
#include <hip/hip_runtime.h>

typedef __attribute__((ext_vector_type(2))) float v2f;
typedef __attribute__((ext_vector_type(4))) float v4f;
typedef __attribute__((ext_vector_type(8))) float v8f;

#define NROWS   16384        // B*L
#define DDIM    320
#define CDIM    16
#define KAUG    20           // 16 dims + [norm, 1] + 2 zero pad -> 5 k-slices of 4
#define KCODES  8192
#define KTILES  (KCODES / 16)      // 512
#define ROWTILES (NROWS / 16)      // 1024
#define CHUNKS  4
#define TILES_PER_CHUNK (KTILES / CHUNKS)  // 128

// ---------------- kernel 1: init best key ----------------
__global__ void rpq_init_kernel(unsigned long long* best) {
    if (threadIdx.x == 0 && blockIdx.x == 0) *best = ~0ull;
}

// ---------------- kernel 2: projection -> augmented A rows ----------------
// One block = 16 rows. LDS-staged X rows and W (stride padded 320->321:
// 321 % 64 == 1 -> conflict-free across the 64 LDS banks).
// Emits A'[row] = [-2*p0..-2*p15, (mask ? ||p||^2 : +inf), 1, 0, 0]
// so the distance GEMM computes sq = ||p||^2 + ||c||^2 - 2 p.c directly,
// with the reference's jnp.where(mask, ., inf) baked into the operand.
__global__ void rpq_proj_kernel(const float* __restrict__ x,
                                const float* __restrict__ w,
                                const int*   __restrict__ mask,
                                float* __restrict__ proj_aug) {
    __shared__ float Xs[16 * 321];
    __shared__ float Ws[16 * 321];
    __shared__ float sqs[256];

    const int tid  = threadIdx.x;
    const int row0 = blockIdx.x * 16;

    for (int i = tid; i < 16 * DDIM; i += 256) {
        int r = i / DDIM, d = i - r * DDIM;
        Ws[r * 321 + d] = w[i];
        Xs[r * 321 + d] = x[(row0 + r) * DDIM + d];
    }
    __syncthreads();

    const int r = tid >> 4;       // 0..15
    const int c = tid & 15;       // 0..15
    float acc = 0.f;
    #pragma unroll 4
    for (int d = 0; d < DDIM; ++d)
        acc = fmaf(Xs[r * 321 + d], Ws[c * 321 + d], acc);

    proj_aug[(row0 + r) * KAUG + c] = -2.f * acc;

    sqs[tid] = acc * acc;
    __syncthreads();
    if (tid < 16) {
        float s = 0.f;
        #pragma unroll
        for (int j = 0; j < 16; ++j) s += sqs[tid * 16 + j];
        const int row = row0 + tid;
        float* dst = proj_aug + row * KAUG;
        dst[16] = (mask[row] == 1) ? s : __builtin_inff();
        dst[17] = 1.f;
        dst[18] = 0.f;
        dst[19] = 0.f;
    }
}

// ---------------- kernel 3: codebook -> augmented B columns ----------------
// B'[code] = [c0..c15, 1, ||c||^2, 0, 0]
__global__ void rpq_cbaug_kernel(const float* __restrict__ cb,
                                 float* __restrict__ cb_aug) {
    const int k = blockIdx.x * blockDim.x + threadIdx.x;
    if (k >= KCODES) return;
    const v4f* p = (const v4f*)(cb + k * CDIM);
    v4f* dst = (v4f*)(cb_aug + k * KAUG);       // k*20 floats = 80B -> 16B aligned
    float s = 0.f;
    #pragma unroll
    for (int q = 0; q < 4; ++q) {
        v4f v = p[q];
        s += v.x * v.x + v.y * v.y + v.z * v.z + v.w * v.w;
        dst[q] = v;
    }
    v4f tail = {1.f, s, 0.f, 0.f};
    dst[4] = tail;
}

// ---------------- kernel 4: fused WMMA distance + argmin ----------------
// Each wave: one 16-row tile x 128 code tiles. sq(16x16) = A'(16x20) * B'(20x16)
// via 5 chained v_wmma_f32_16x16x4_f32 (full f32, matches reference numerics;
// norms, -2 scale, and inf-masking are inside the operands). Per-element
// epilogue is only v_max_num_f32 + key build + v_min_u64.
// Lexicographic (value, flat_idx) min via u64 key -> global_atomic_min_u64.
__global__ void rpq_dist_argmin_kernel(const float* __restrict__ proj_aug,
                                       const float* __restrict__ cb_aug,
                                       unsigned long long* __restrict__ best) {
    const int lane = threadIdx.x & 31;
    const int wid  = blockIdx.x * (blockDim.x >> 5) + (threadIdx.x >> 5);
    const int row_tile = wid >> 2;            // 0..1023
    const int chunk    = wid & 3;             // 0..3
    const int row0     = row_tile << 4;

    const int m16  = lane & 15;
    const int half = lane >> 4;               // 0: K0/K1 lanes, 1: K2/K3 lanes

    // A-matrix (16x4 f32) fragments for the 5 k-slices of KAUG=20.
    const float* pr = proj_aug + (row0 + m16) * KAUG + half * 2;
    v2f a0 = *(const v2f*)(pr + 0);
    v2f a1 = *(const v2f*)(pr + 4);
    v2f a2 = *(const v2f*)(pr + 8);
    v2f a3 = *(const v2f*)(pr + 12);
    v2f a4 = *(const v2f*)(pr + 16);

    // This lane's 8 output rows: M = j + (half ? 8 : 0).
    const int mbase = row0 + half * 8;
    unsigned long long rowkey[8];             // (row flat base) part of the key
    #pragma unroll
    for (int j = 0; j < 8; ++j)
        rowkey[j] = (unsigned long long)((unsigned)(mbase + j) << 13);  // *KCODES

    unsigned long long bestkey = ~0ull;
    const int ct0 = chunk * TILES_PER_CHUNK;

    #pragma unroll 2
    for (int ct = ct0; ct < ct0 + TILES_PER_CHUNK; ++ct) {
        const int code0 = ct << 4;
        const int n = m16;                    // output column within tile
        const float* cbp = cb_aug + (code0 + n) * KAUG + half * 2;

        // prefetch a few B tiles ahead (emits global_prefetch_b8)
        __builtin_prefetch(cbp + 4 * 16 * KAUG, 0, 1);

        v2f b0 = *(const v2f*)(cbp + 0);
        v2f b1 = *(const v2f*)(cbp + 4);
        v2f b2 = *(const v2f*)(cbp + 8);
        v2f b3 = *(const v2f*)(cbp + 12);
        v2f b4 = *(const v2f*)(cbp + 16);

        v8f acc = {};
        acc = __builtin_amdgcn_wmma_f32_16x16x4_f32(false, a0, false, b0,
                                                    (short)0, acc, false, false);
        acc = __builtin_amdgcn_wmma_f32_16x16x4_f32(false, a1, false, b1,
                                                    (short)0, acc, false, false);
        acc = __builtin_amdgcn_wmma_f32_16x16x4_f32(false, a2, false, b2,
                                                    (short)0, acc, false, false);
        acc = __builtin_amdgcn_wmma_f32_16x16x4_f32(false, a3, false, b3,
                                                    (short)0, acc, false, false);
        acc = __builtin_amdgcn_wmma_f32_16x16x4_f32(false, a4, false, b4,
                                                    (short)0, acc, false, false);

        const unsigned colidx = (unsigned)(code0 + n);
        #pragma unroll
        for (int j = 0; j < 8; ++j) {
            float sq = fmaxf(acc[j], 0.f);    // acc already = pn + cn - 2*dot (inf if unmasked)
            unsigned long long key =
                ((unsigned long long)__float_as_uint(sq) << 32) |
                (rowkey[j] + colidx);
            bestkey = key < bestkey ? key : bestkey;
        }
    }

    // wave32 shuffle reduction of the u64 key
    #pragma unroll
    for (int off = 16; off > 0; off >>= 1) {
        unsigned long long o = __shfl_down(bestkey, off, 32);
        bestkey = o < bestkey ? o : bestkey;
    }
    if (lane == 0) atomicMin(best, bestkey);
}

// ---------------- kernel 5: finalize (mask prefix count + output) ----------------
__global__ void rpq_finalize_kernel(const unsigned long long* __restrict__ best,
                                    const int* __restrict__ mask,
                                    float* __restrict__ out) {
    __shared__ int part[256];
    const unsigned long long key = *best;
    const unsigned idx = (unsigned)(key & 0xFFFFFFFFu);
    const int r = (int)(idx >> 13);           // idx / 8192
    const int c = (int)(idx & (KCODES - 1));  // idx % 8192

    const int tid = threadIdx.x;
    int s = 0;
    for (int i = tid; i <= r; i += 256) s += (mask[i] == 1);
    part[tid] = s;
    __syncthreads();
    #pragma unroll
    for (int off = 128; off > 0; off >>= 1) {
        if (tid < off) part[tid] += part[tid + off];
        __syncthreads();
    }
    if (tid == 0) {
        long long compact = (long long)part[0] - 1;
        out[0] = (float)(compact * (long long)KCODES + (long long)c);
    }
}

// ---------------- launch ----------------
extern "C" void kernel_launch(void* const* d_in, const int* in_sizes, int n_in,
                              void* d_out, int out_size, void* d_ws, size_t ws_size,
                              hipStream_t stream) {
    (void)in_sizes; (void)n_in; (void)out_size; (void)ws_size;

    const float* x    = (const float*)d_in[0];   // (8,2048,320) f32
    const int*   mask = (const int*)  d_in[1];   // (8,2048) i32
    const float* w    = (const float*)d_in[2];   // (16,320) f32
    const float* cb   = (const float*)d_in[3];   // (8192,16) f32
    float* out = (float*)d_out;

    char* ws = (char*)d_ws;
    unsigned long long* best = (unsigned long long*)ws;                 // 8 B
    float* proj_aug = (float*)(ws + 1024);                              // 16384*20*4 = 1.25 MB
    float* cb_aug   = (float*)(ws + 1024 + NROWS * KAUG * 4);           // 8192*20*4  = 640 KB

    rpq_init_kernel<<<1, 32, 0, stream>>>(best);
    rpq_proj_kernel<<<ROWTILES, 256, 0, stream>>>(x, w, mask, proj_aug);
    rpq_cbaug_kernel<<<KCODES / 256, 256, 0, stream>>>(cb, cb_aug);
    // 4096 waves = 1024 row tiles x 4 K-chunks; 8 waves per 256-thread block
    rpq_dist_argmin_kernel<<<512, 256, 0, stream>>>(proj_aug, cb_aug, best);
    rpq_finalize_kernel<<<1, 256, 0, stream>>>(best, mask, out);
}